// VectorQuantizer_39960375722359
// MI455X (gfx1250) — compile-verified
//
#include <hip/hip_runtime.h>
#include <hip/hip_bf16.h>

typedef float v2f __attribute__((ext_vector_type(2)));
typedef float v8f __attribute__((ext_vector_type(8)));
typedef int   v4i __attribute__((ext_vector_type(4)));

#define EMB_DIM 256
#define KCODES 1024
#define NTILES (KCODES / 16)    // 64 code tiles of 16 codes each
#define KSTEPS (EMB_DIM / 4)    // 64 wmma k-steps (K=4 per wmma)
#define ROWS_PER_WAVE 16
#define WAVES_PER_BLOCK 8
#define ROWS_PER_BLOCK (ROWS_PER_WAVE * WAVES_PER_BLOCK)  // 128
#define TOTAL_ROWS 32768        // 8 * 4096
#define TILE_FLOATS (16 * EMB_DIM)          // 4096 floats = 16 KB
#define TILE_VEC4 (TILE_FLOATS / 4)         // 1024 float4
#define VEC4_PER_THREAD (TILE_VEC4 / 256)   // 4 async b128 loads per thread per tile

// ---- gfx1250 async global->LDS path (ASYNCcnt), with safe fallback ----
#if __has_builtin(__builtin_amdgcn_global_load_async_to_lds_b128)
#define VQ_ASYNC 1
#else
#define VQ_ASYNC 0
#endif

#if __has_builtin(__builtin_amdgcn_s_wait_asynccnt)
#define WAIT_ASYNCCNT(n) __builtin_amdgcn_s_wait_asynccnt(n)
#else
#define WAIT_ASYNCCNT(n) asm volatile("s_wait_asynccnt %0" ::"n"(n) : "memory")
#endif

#if VQ_ASYNC
// Builtin signature (from clang diagnostic): (global v4i*, local v4i*, imm offset, imm cpol)
#define ASYNC_B128(gsrc, ldst)                                              \
    __builtin_amdgcn_global_load_async_to_lds_b128(                         \
        (__attribute__((address_space(1))) v4i*)(gsrc),                     \
        (__attribute__((address_space(3))) v4i*)(ldst), 0, 0)
#endif

__launch_bounds__(256, 1)
__global__ void vq_wmma_kernel(const float* __restrict__ x,
                               const float* __restrict__ emb,
                               float* __restrict__ out_st,   // x + (e - x)
                               float* __restrict__ out_emb)  // e
{
    __shared__ float ebuf[2][TILE_FLOATS];        // 32 KB: double-buffered code tile
    __shared__ int   bestIdx[ROWS_PER_BLOCK];     // per-row winning code

    const int tid    = threadIdx.x;
    const int lane   = tid & 31;
    const int wave   = tid >> 5;
    const int lane15 = lane & 15;
    const int hi     = lane >> 4;      // 0: K=k0,k0+1 half | 1: K=k0+2,k0+3 half
    const int koff   = hi * 2;

    const int rowBase = blockIdx.x * ROWS_PER_BLOCK + wave * ROWS_PER_WAVE;
    const int myRow   = rowBase + lane15;

    // ---- Load the wave's A tile (16 rows x 256 dims) into registers ----
    // ISA 16x4 f32 A layout: lanes 0-15 hold (K=0,1) in v0,v1; lanes 16-31 hold (K=2,3).
    v2f a[KSTEPS];
    {
        const float* xrow = x + (size_t)myRow * EMB_DIM + koff;
#pragma unroll
        for (int kk = 0; kk < KSTEPS; ++kk)
            a[kk] = *(const v2f*)(xrow + kk * 4);
    }

    float vmin[8];
    int   vtile[8];
#pragma unroll
    for (int g = 0; g < 8; ++g) { vmin[g] = 3.0e38f; vtile[g] = 0; }

#if VQ_ASYNC
    // Prologue: launch tile 0 into buffer 0 (4 x b128 per thread, ASYNCcnt += 4)
    {
        const float4* src = (const float4*)emb;
        float4* dst = (float4*)ebuf[0];
#pragma unroll
        for (int i = 0; i < VEC4_PER_THREAD; ++i)
            ASYNC_B128(src + tid + 256 * i, dst + tid + 256 * i);
    }
#endif

    for (int tile = 0; tile < NTILES; ++tile) {
#if VQ_ASYNC
        // All waves must be done reading the buffer we are about to overwrite
        // (it held tile-1's data) before issuing tile+1's async loads into it.
        __syncthreads();
        if (tile + 1 < NTILES) {
            const float4* src = (const float4*)(emb + (size_t)(tile + 1) * TILE_FLOATS);
            float4* dst = (float4*)ebuf[(tile + 1) & 1];
#pragma unroll
            for (int i = 0; i < VEC4_PER_THREAD; ++i)
                ASYNC_B128(src + tid + 256 * i, dst + tid + 256 * i);
            // 8 outstanding; async loads complete in order -> waiting to <=4
            // guarantees tile's 4 landed while tile+1's stay in flight.
            WAIT_ASYNCCNT(VEC4_PER_THREAD);
        } else {
            WAIT_ASYNCCNT(0);
        }
        __syncthreads();
        const float* tbase = ebuf[tile & 1];
#else
        // Synchronous fallback staging
        __syncthreads();
        {
            const float4* src = (const float4*)(emb + (size_t)tile * TILE_FLOATS);
            float4* dst = (float4*)ebuf[0];
#pragma unroll
            for (int i = 0; i < VEC4_PER_THREAD; ++i)
                dst[tid + 256 * i] = src[tid + 256 * i];
        }
        __syncthreads();
        if (tile + 1 < NTILES)
            __builtin_prefetch(emb + (size_t)(tile + 1) * TILE_FLOATS + tid * 16, 0, 0);
        const float* tbase = ebuf[0];
#endif

        // ---- 16x16 tile of S = X . E^T via 64 fp32 WMMAs; fuse ||e||^2 ----
        v8f c = {0.f, 0.f, 0.f, 0.f, 0.f, 0.f, 0.f, 0.f};
        float s = 0.f;
        const float* ebase = tbase + lane15 * EMB_DIM + koff;  // B[k][n]=emb[n][k]
#pragma unroll
        for (int kk = 0; kk < KSTEPS; ++kk) {
            v2f b = *(const v2f*)(ebase + kk * 4);
            s += b.x * b.x + b.y * b.y;
            c = __builtin_amdgcn_wmma_f32_16x16x4_f32(
                    false, a[kk], false, b, (short)0, c, false, false);
        }
        // both lane halves of a code hold complementary K subsets
        const float e2 = s + __shfl_xor(s, 16, 32);

        // dist[m][n] = e2[n] - 2*S[m][n]; track running min per (C-vgpr, lane)
#pragma unroll
        for (int g = 0; g < 8; ++g) {
            float d = e2 - 2.0f * c[g];
            if (d < vmin[g]) { vmin[g] = d; vtile[g] = tile; }
        }
    }

    // ---- Argmin across the 16 lanes sharing each row (first-index tie-break) ----
#pragma unroll
    for (int g = 0; g < 8; ++g) {
        float m    = vmin[g];
        int   code = vtile[g] * 16 + lane15;
#pragma unroll
        for (int off = 8; off >= 1; off >>= 1) {
            float om = __shfl_xor(m, off, 32);
            int   oc = __shfl_xor(code, off, 32);
            if (om < m || (om == m && oc < code)) { m = om; code = oc; }
        }
        // lanes 0-15 carry row g; lanes 16-31 carry row g+8
        if (lane15 == 0) bestIdx[wave * ROWS_PER_WAVE + hi * 8 + g] = code;
    }
    __syncthreads();

    // ---- Cooperative gather + straight-through output ----
    // 128 rows x 64 float4 chunks = 8192 chunks; 32 per thread, coalesced.
    for (int i = 0; i < 32; ++i) {
        const int chunk = tid + 256 * i;
        const int r     = chunk >> 6;          // row within block
        const int o     = (chunk & 63) * 4;    // float offset within row
        const int row   = blockIdx.x * ROWS_PER_BLOCK + r;
        const int bi    = bestIdx[r];

        const float4 e  = *(const float4*)(emb + (size_t)bi * EMB_DIM + o);
        const float4 xv = *(const float4*)(x + (size_t)row * EMB_DIM + o);
        float4 ov;
        ov.x = xv.x + (e.x - xv.x);   // replicate reference fp32 arithmetic exactly
        ov.y = xv.y + (e.y - xv.y);
        ov.z = xv.z + (e.z - xv.z);
        ov.w = xv.w + (e.w - xv.w);

        *(float4*)(out_st  + (size_t)row * EMB_DIM + o) = ov;
        *(float4*)(out_emb + (size_t)row * EMB_DIM + o) = e;
    }
}

extern "C" void kernel_launch(void* const* d_in, const int* in_sizes, int n_in,
                              void* d_out, int out_size, void* d_ws, size_t ws_size,
                              hipStream_t stream) {
    (void)in_sizes; (void)n_in; (void)d_ws; (void)ws_size; (void)out_size;
    const float* x   = (const float*)d_in[0];   // [8,4096,256] fp32
    const float* emb = (const float*)d_in[1];   // [1024,256]  fp32
    float* out_st  = (float*)d_out;                                   // first tuple element
    float* out_emb = (float*)d_out + (size_t)TOTAL_ROWS * EMB_DIM;    // second tuple element

    dim3 grid(TOTAL_ROWS / ROWS_PER_BLOCK);  // 256 blocks
    dim3 block(256);                          // 8 wave32s
    vq_wmma_kernel<<<grid, block, 0, stream>>>(x, emb, out_st, out_emb);
}